// AsymmetricLossCustomPrioritySmallFocal_18064632447147
// MI455X (gfx1250) — compile-verified
//
#include <hip/hip_runtime.h>
#include <hip/hip_bf16.h>
#include <stdint.h>

// AsymmetricLossCustomPrioritySmallFocal for MI455X (gfx1250).
// Memory-bound: 157MB of x/y per call -> ~6.7us floor at 23.3 TB/s.
// Strategy: one workgroup per row; stage the full x/y row pair (77KB) into
// LDS via CDNA5 async global->LDS DMA (ASYNCcnt), then run both passes
// (top-10 threshold + whitelist flags, then fused loss) out of LDS so HBM
// sees each byte exactly once.

namespace {

constexpr int   kC    = 9605;   // classes (reference C)
constexpr int   kTPB  = 256;    // 8 x wave32
constexpr int   kTopK = 10;
constexpr int   kLdsN = 9616;   // kC + alignment shift (<=3), padded

constexpr float kClip  = 0.05f;
constexpr float kEps   = 1e-8f;
constexpr float kAlpha = 2.0f;

#define AS1 __attribute__((address_space(1)))
#define AS3 __attribute__((address_space(3)))

typedef int v4i __attribute__((ext_vector_type(4)));

#if defined(__HIP_DEVICE_COMPILE__) &&                                  \
    __has_builtin(__builtin_amdgcn_global_load_async_to_lds_b128) &&    \
    __has_builtin(__builtin_amdgcn_global_load_async_to_lds_b32)
#define HAVE_ASYNC_LDS 1
#endif

#if defined(HAVE_ASYNC_LDS)
// Builtin signature (from clang diagnostics): (AS1 T*, AS3 T*, imm, imm),
// T = int for b32, T = int4 for b128. Pointers are per-lane.
__device__ __forceinline__ void async_copy_b128(const float* g, float* l) {
  __builtin_amdgcn_global_load_async_to_lds_b128(
      (AS1 v4i*)(g), (AS3 v4i*)(l), 0, 0);
}
__device__ __forceinline__ void async_copy_b32(const float* g, float* l) {
  __builtin_amdgcn_global_load_async_to_lds_b32(
      (AS1 int*)(g), (AS3 int*)(l), 0, 0);
}
#endif

// Unrolled descending top-k insert: constant indices only -> stays in VGPRs.
__device__ __forceinline__ void topk_insert(float (&t)[kTopK], float v) {
  if (v <= t[kTopK - 1]) return;
#pragma unroll
  for (int j = 0; j < kTopK; ++j) {
    const float cur = t[j];
    const bool  gt  = v > cur;
    t[j] = gt ? v : cur;
    v    = gt ? cur : v;
  }
}

__global__ __launch_bounds__(kTPB)
void asl_row_kernel(const float* __restrict__ x, const float* __restrict__ y,
                    const int* __restrict__ cat,
                    const unsigned char* __restrict__ in_mapping,
                    float* __restrict__ partial) {
  __shared__ alignas(16) float lx[kLdsN];
  __shared__ alignas(16) float ly[kLdsN];
  __shared__ float ltop[kTPB * kTopK];
  __shared__ float red[kTPB];
  __shared__ int   flags[3];

  const int row = blockIdx.x;
  const int tid = threadIdx.x;
  const float* __restrict__ xr = x + (size_t)row * kC;
  const float* __restrict__ yr = y + (size_t)row * kC;

  // x and y rows share the same misalignment (identical element offset,
  // 256B-aligned hipMalloc bases). Shift the LDS image by a0 elements so
  // every b128 transfer is 16B-aligned on both the global and LDS side.
  const int a0    = (int)(((uintptr_t)xr >> 2) & 3);
  const int head  = (4 - a0) & 3;
  const int nfull = (kC - head) >> 2;
  const int tailc = head + (nfull << 2);
  const int ntail = kC - tailc;

#if defined(HAVE_ASYNC_LDS)
  if (tid < head) {
    async_copy_b32(xr + tid, &lx[a0 + tid]);
    async_copy_b32(yr + tid, &ly[a0 + tid]);
  }
  for (int k = tid; k < nfull; k += kTPB) {
    const int c = head + (k << 2);
    async_copy_b128(xr + c, &lx[a0 + c]);
    async_copy_b128(yr + c, &ly[a0 + c]);
  }
  if (tid < ntail) {
    const int c = tailc + tid;
    async_copy_b32(xr + c, &lx[a0 + c]);
    async_copy_b32(yr + c, &ly[a0 + c]);
  }
#if __has_builtin(__builtin_amdgcn_s_wait_asynccnt)
  __builtin_amdgcn_s_wait_asynccnt(0);
#else
  asm volatile("s_wait_asynccnt 0" ::: "memory");
#endif
#else
  // Fallback staging path (no async-to-LDS builtins in this toolchain).
  for (int c = tid; c < kC; c += kTPB) {
    lx[a0 + c] = xr[c];
    ly[a0 + c] = yr[c];
  }
#endif
  if (tid < 3) flags[tid] = 0;
  __syncthreads();

  // ---------------- Pass 1: whitelist flags + per-thread top-10 ----------
  float t[kTopK];
#pragma unroll
  for (int j = 0; j < kTopK; ++j) t[j] = -__builtin_huge_valf();

  int f1 = 0, f2 = 0, f3 = 0;
  for (int c = tid; c < kC; c += kTPB) {
    topk_insert(t, lx[a0 + c]);
    if (ly[a0 + c] != 0.0f) {          // positives are ~1%: cat read is rare
      const int cc = cat[c];
      f1 |= (cc == 1);
      f2 |= (cc == 2);
      f3 |= (cc == 3);
    }
  }
  if (f1) atomicOr(&flags[0], 1);
  if (f2) atomicOr(&flags[1], 1);
  if (f3) atomicOr(&flags[2], 1);

#pragma unroll
  for (int j = 0; j < kTopK; ++j) ltop[tid * kTopK + j] = t[j];
  __syncthreads();

  // Tree-merge 256 sorted top-10 lists down to one.
  for (int s = kTPB / 2; s >= 1; s >>= 1) {
    if (tid < s) {
#pragma unroll
      for (int j = 0; j < kTopK; ++j)
        topk_insert(t, ltop[(tid + s) * kTopK + j]);
#pragma unroll
      for (int j = 0; j < kTopK; ++j) ltop[tid * kTopK + j] = t[j];
    }
    __syncthreads();
  }

  const float thresh = ltop[kTopK - 1];   // 10th largest x (sigmoid monotone)
  const bool  p1   = flags[0] != 0;
  const bool  p2   = flags[1] != 0;
  const bool  p3   = flags[2] != 0;
  const bool  has4 = !(p1 | p2 | p3);

  // ---------------- Pass 2: fused loss -----------------------------------
  float acc = 0.0f;
  for (int c = tid; c < kC; c += kTPB) {
    const float xv = lx[a0 + c];
    const float yy = ly[a0 + c];
    const float xs_pos = 1.0f / (1.0f + __expf(-xv));
    const float xs_neg = fminf(1.0f - xs_pos + kClip, 1.0f);
    const int   cc = cat[c];
    const bool  im = in_mapping[c] != 0;

    float loss, w, factor;
    if (yy != 0.0f) {                       // y == 1: gamma = 1
      loss   = __logf(fmaxf(xs_pos, kEps));
      w      = 1.0f - xs_pos;               // (1-pt)^1
      factor = xs_neg * kAlpha;
    } else {                                // y == 0: gamma = 4
      loss   = __logf(fmaxf(xs_neg, kEps));
      float u = 1.0f - xs_neg;
      u = u * u;
      w = u * u;                            // (1-pt)^4
      factor = xs_pos * kAlpha;
    }
    const bool condA = (!im) && has4;
    const bool condB = (cc == 1 && p1) || (cc == 2 && p2) ||
                       (cc == 3 && p3) || (cc == 4 && has4);
    if ((xv >= thresh) && (condA || condB)) loss *= factor;
    acc += loss * w;
  }

  // Deterministic block reduction -> one partial per row.
  red[tid] = acc;
  __syncthreads();
  for (int s = kTPB / 2; s >= 1; s >>= 1) {
    if (tid < s) red[tid] += red[tid + s];
    __syncthreads();
  }
  if (tid == 0) partial[row] = red[0];
}

__global__ __launch_bounds__(kTPB)
void asl_reduce_kernel(const float* __restrict__ partial, int n,
                       float* __restrict__ out) {
  __shared__ float red[kTPB];
  float a = 0.0f;
  for (int i = threadIdx.x; i < n; i += kTPB) a += partial[i];
  red[threadIdx.x] = a;
  __syncthreads();
  for (int s = kTPB / 2; s >= 1; s >>= 1) {
    if (threadIdx.x < s) red[threadIdx.x] += red[threadIdx.x + s];
    __syncthreads();
  }
  if (threadIdx.x == 0) out[0] = -red[0];
}

}  // namespace

extern "C" void kernel_launch(void* const* d_in, const int* in_sizes, int n_in,
                              void* d_out, int out_size, void* d_ws,
                              size_t ws_size, hipStream_t stream) {
  (void)n_in; (void)out_size; (void)ws_size;
  const float*         x      = (const float*)d_in[0];
  const float*         y      = (const float*)d_in[1];
  const int*           cat    = (const int*)d_in[2];
  const unsigned char* inmap  = (const unsigned char*)d_in[3];  // jax bool = u8
  const int C = in_sizes[2] > 0 ? in_sizes[2] : kC;
  const int B = in_sizes[0] / C;

  float* partial = (float*)d_ws;          // B floats of scratch
  asl_row_kernel<<<B, kTPB, 0, stream>>>(x, y, cat, inmap, partial);
  asl_reduce_kernel<<<1, kTPB, 0, stream>>>(partial, B, (float*)d_out);
}